// ImpGraphConvolution_15015205667141
// MI455X (gfx1250) — compile-verified
//
#include <hip/hip_runtime.h>
#include <hip/hip_bf16.h>

#define N_NODES 100000
#define N_EDGES 1600000
#define D 64
#define TILES_M (N_NODES / 16)   // 6250, exact
#define WSP_STRIDE 160           // floats per k-pair row; 160 % 64 == 32 dwords
                                 // -> half-waves hit disjoint LDS bank halves

typedef __attribute__((ext_vector_type(2))) float v2f;
typedef __attribute__((ext_vector_type(8))) float v8f;

// ---------------------------------------------------------------------------
// out[n][d] = bias[d]  (SPMM atomics accumulate on top of this)
// ---------------------------------------------------------------------------
__global__ void init_out_kernel(float* __restrict__ out,
                                const float* __restrict__ bias) {
    int i = blockIdx.x * blockDim.x + threadIdx.x;
    if (i < N_NODES * D) out[i] = bias[i & (D - 1)];
}

// ---------------------------------------------------------------------------
// support = x @ (W_own + W_nbr + W_temp), fp32 WMMA 16x16x4.
// B is pre-packed in LDS as K-pairs: WsP[q][n] = {W[2q][n], W[2q+1][n]}
// so each B fragment is ONE aligned ds_load_b64 (no 2addr + register shuffle).
// ---------------------------------------------------------------------------
__global__ void __launch_bounds__(256)
gemm_kernel(const float* __restrict__ x,
            const float* __restrict__ w0,
            const float* __restrict__ w1,
            const float* __restrict__ w2,
            float* __restrict__ support) {
    __shared__ float WsP[(D / 2) * WSP_STRIDE];  // 32 k-pairs x 160 = 20 KB

    int tid = threadIdx.x;
    for (int i = tid; i < D * D; i += 256) {
        int r = i >> 6, c = i & 63;
        // pack element (r,c) into pair slot (q = r/2, component r&1)
        WsP[(r >> 1) * WSP_STRIDE + c * 2 + (r & 1)] = w0[i] + w1[i] + w2[i];
    }
    __syncthreads();

    int wave = tid >> 5;
    int lane = tid & 31;
    int tile = blockIdx.x * 8 + wave;          // per-wave uniform
    if (tile >= TILES_M) return;               // whole-wave exit: EXEC stays all-1s

    int row0  = tile * 16;
    int m     = lane & 15;                     // M (A) / N (B,C) position within 16
    int khalf = (lane >> 4) * 2;               // lanes 0-15 -> K {k,k+1}; 16-31 -> {k+2,k+3}
    int hi8   = (lane >> 4) * 8;               // C/D: upper lanes hold rows M+8

    v8f acc0 = {}, acc1 = {}, acc2 = {}, acc3 = {};
    const float* arow = x + (size_t)(row0 + m) * D;

#pragma unroll
    for (int kk = 0; kk < D; kk += 4) {
        // A fragment: 32-bit A 16x4 layout -> contiguous float2 per lane
        v2f a = *(const v2f*)(arow + kk + khalf);

        int q = (kk + khalf) >> 1;             // k-pair index for this half-wave
        const float* brow = WsP + q * WSP_STRIDE + m * 2;

        v2f b0 = *(const v2f*)(brow +  0);     // n-tile 0: cols  m
        v2f b1 = *(const v2f*)(brow + 32);     // n-tile 1: cols 16+m
        v2f b2 = *(const v2f*)(brow + 64);     // n-tile 2: cols 32+m
        v2f b3 = *(const v2f*)(brow + 96);     // n-tile 3: cols 48+m

        acc0 = __builtin_amdgcn_wmma_f32_16x16x4_f32(false, a, false, b0,
                                                     (short)0, acc0, false, false);
        acc1 = __builtin_amdgcn_wmma_f32_16x16x4_f32(false, a, false, b1,
                                                     (short)0, acc1, false, false);
        acc2 = __builtin_amdgcn_wmma_f32_16x16x4_f32(false, a, false, b2,
                                                     (short)0, acc2, false, false);
        acc3 = __builtin_amdgcn_wmma_f32_16x16x4_f32(false, a, false, b3,
                                                     (short)0, acc3, false, false);
    }

    // C/D layout: VGPR i -> M = i (+8 for lanes 16-31), N = lane%16
#pragma unroll
    for (int i = 0; i < 8; ++i) {
        size_t r = (size_t)(row0 + i + hi8) * D;
        support[r +  0 + m] = acc0[i];
        support[r + 16 + m] = acc1[i];
        support[r + 32 + m] = acc2[i];
        support[r + 48 + m] = acc3[i];
    }
}

// ---------------------------------------------------------------------------
// SPMM: one wave per edge. out[row] += val * support[col]  (64 floats/edge).
// support & out both fit in 192MB L2, so gather + atomics stay on-chip.
// ---------------------------------------------------------------------------
__global__ void __launch_bounds__(256)
spmm_kernel(const int* __restrict__ rows,
            const int* __restrict__ cols,
            const float* __restrict__ vals,
            const float* __restrict__ support,
            float* __restrict__ out) {
    int e    = (blockIdx.x * blockDim.x + threadIdx.x) >> 5;
    int lane = threadIdx.x & 31;
    if (e >= N_EDGES) return;

    int   r = rows[e];     // wave-uniform -> scalar path
    int   c = cols[e];
    float v = vals[e];

    const float* src = support + (size_t)c * D;
    float*       dst = out     + (size_t)r * D;

    v2f s = *(const v2f*)(src + lane * 2);     // coalesced 256B per wave
    unsafeAtomicAdd(dst + lane * 2 + 0, v * s.x);
    unsafeAtomicAdd(dst + lane * 2 + 1, v * s.y);
}

// ---------------------------------------------------------------------------
extern "C" void kernel_launch(void* const* d_in, const int* in_sizes, int n_in,
                              void* d_out, int out_size, void* d_ws, size_t ws_size,
                              hipStream_t stream) {
    const float* x        = (const float*)d_in[0];
    const int*   rows     = (const int*)  d_in[1];
    const int*   cols     = (const int*)  d_in[2];
    const float* vals     = (const float*)d_in[3];
    const float* w_own    = (const float*)d_in[4];
    const float* w_nbr    = (const float*)d_in[5];
    const float* w_temp   = (const float*)d_in[6];
    const float* bias     = (const float*)d_in[7];
    float*       out      = (float*)d_out;
    float*       support  = (float*)d_ws;      // 100000*64*4 = 25.6 MB

    // 1) out = bias (broadcast)
    init_out_kernel<<<(N_NODES * D + 255) / 256, 256, 0, stream>>>(out, bias);

    // 2) support = x @ (W0+W1+W2) via fp32 WMMA
    int gemm_blocks = (TILES_M + 7) / 8;       // 8 waves (16-row tiles) per block
    gemm_kernel<<<gemm_blocks, 256, 0, stream>>>(x, w_own, w_nbr, w_temp, support);

    // 3) out += spmm(edges, support)
    int spmm_blocks = (N_EDGES + 7) / 8;       // 8 edges (waves) per block
    spmm_kernel<<<spmm_blocks, 256, 0, stream>>>(rows, cols, vals, support, out);
}